// Counter_66408784331277
// MI455X (gfx1250) — compile-verified
//
#include <hip/hip_runtime.h>

typedef float v2f __attribute__((ext_vector_type(2)));
typedef float v4f __attribute__((ext_vector_type(4)));
typedef float v8f __attribute__((ext_vector_type(8)));

#define IN_H 512
#define IN_W 512
#define OUT_D 127
#define H_ROWS 68          // input rows needed for 16 output rows: 4*15+8 = 68
#define H_STRIDE 144       // padded LDS row stride (floats): 2*144 % 64 == 32 -> no bank conflicts

// grid = (8 row-tiles, 256 planes), block = 256 threads (8 wave32)
__global__ __launch_bounds__(256) void boxsum8x8s4_kernel(const float* __restrict__ x,
                                                          float* __restrict__ out) {
    const int ty    = blockIdx.x;   // output row tile (16 rows each, 8 tiles cover 127)
    const int plane = blockIdx.y;   // n*c plane, 0..255
    const float* __restrict__ xp = x + (size_t)plane * (IN_H * IN_W);

    __shared__ float Hs[H_ROWS * H_STRIDE];

    const int t    = threadIdx.x;
    const int row0 = 64 * ty;       // first input row of this band

    // ---------------- Phase 1: horizontal pass into LDS ----------------
    // H[r][j] = sum_{u<8} x[row0+r][4j+u], computed as pairs sharing the middle float4.
    // 68 rows * 64 pairs = 4352 pairs; 256 threads * 17 iters.
#pragma unroll
    for (int it = 0; it < 17; ++it) {
        const int e  = t + it * 256;   // 0..4351
        const int r  = e >> 6;         // 0..67
        const int jp = e & 63;         // pair index -> cols 2*jp, 2*jp+1
        const int gr = row0 + r;
        float h0 = 0.0f, h1 = 0.0f;
        if (gr < IN_H) {
            const float* rowp = xp + gr * IN_W + 8 * jp;    // 16B aligned
            v4f A = *(const v4f*)(rowp);
            v4f B = *(const v4f*)(rowp + 4);
            float sA = (A.x + A.y) + (A.z + A.w);
            float sB = (B.x + B.y) + (B.z + B.w);
            float sC = 0.0f;
            if (jp < 63) {                                   // jp==63 tail never stored
                v4f C = *(const v4f*)(rowp + 8);
                sC = (C.x + C.y) + (C.z + C.w);
            }
            h0 = sA + sB;   // window cols 8jp .. 8jp+7
            h1 = sB + sC;   // window cols 8jp+4 .. 8jp+11
        }
        Hs[r * H_STRIDE + 2 * jp]     = h0;
        Hs[r * H_STRIDE + 2 * jp + 1] = h1;
    }
    __syncthreads();

    // ---------------- Phase 2: vertical pass via v_wmma_f32_16x16x4_f32 ----------------
    // Y(16x16) = S(16x68, ones band) x H(68x16), chunked K=4 -> 17 WMMAs.
    const int lane  = t & 31;
    const int wave  = t >> 5;            // 0..7, each wave owns 16 output cols
    const int j0    = wave * 16;
    const int n     = lane & 15;         // A-frag M index == B-frag N index for this lane
    const int khalf = (lane >> 4) << 1;  // lanes 0-15 -> K {0,1}; lanes 16-31 -> K {2,3}

    v8f c = {};                          // f32 accumulator, 8 VGPRs
#pragma unroll
    for (int q = 0; q < 17; ++q) {
        const int k0 = 4 * q + khalf;    // local H row of this lane's first element
        // A fragment: ones band S[M,k] = (4M <= k < 4M+8)
        const int d0 = k0 - 4 * n;
        v2f a;
        a.x = (d0 >= 0 && d0 < 8)     ? 1.0f : 0.0f;
        a.y = (d0 + 1 >= 0 && d0 + 1 < 8) ? 1.0f : 0.0f;
        // B fragment: H rows k0, k0+1, col j0+n (bank-conflict-free via H_STRIDE pad)
        v2f b;
        b.x = Hs[k0 * H_STRIDE + j0 + n];
        b.y = Hs[(k0 + 1) * H_STRIDE + j0 + n];
        c = __builtin_amdgcn_wmma_f32_16x16x4_f32(
                /*neg_a=*/false, a, /*neg_b=*/false, b,
                /*c_mod=*/(short)0, c, /*reuse_a=*/false, /*reuse_b=*/false);
    }

    // ---------------- Store 16x16 tile (guarded: 127 not multiple of 16) ----------------
    const int gj    = j0 + n;                          // global output col
    const int rbase = 16 * ty + ((lane >> 4) << 3);    // lanes 16-31 hold M = v+8
    if (gj < OUT_D) {
        float* op = out + (size_t)plane * (OUT_D * OUT_D);
#pragma unroll
        for (int v = 0; v < 8; ++v) {
            const int gi = rbase + v;
            if (gi < OUT_D) op[gi * OUT_D + gj] = c[v];
        }
    }
}

extern "C" void kernel_launch(void* const* d_in, const int* in_sizes, int n_in,
                              void* d_out, int out_size, void* d_ws, size_t ws_size,
                              hipStream_t stream) {
    const float* x = (const float*)d_in[0];
    float* out = (float*)d_out;
    (void)in_sizes; (void)n_in; (void)out_size; (void)d_ws; (void)ws_size;
    dim3 grid(8, 8 * 32, 1);   // 8 row tiles x 256 (N*C) planes
    dim3 block(256, 1, 1);     // 8 wave32
    boxsum8x8s4_kernel<<<grid, block, 0, stream>>>(x, out);
}